// MSGIL_NORM_Loss_20882130993727
// MI455X (gfx1250) — compile-verified
//
#include <hip/hip_runtime.h>
#include <hip/hip_bf16.h>

typedef __attribute__((ext_vector_type(2))) float v2f;
typedef __attribute__((ext_vector_type(8))) float v8f;

#define BATCH   32
#define HDIM    1024
#define WDIM    1024
#define NPIX    (HDIM * WDIM)          // 1,048,576 per sample
#define NBINS   4096
#define CHUNKS  16                      // histogram blocks per sample
#define TILE    32
#define EPSF    1e-8f
#define VALID_T (-1e-8f)

// ---------------------------------------------------------------- zero ws
__global__ void msgil_zero_kernel(float* ws, int n) {
    int i = blockIdx.x * blockDim.x + threadIdx.x;
    if (i < n) ws[i] = 0.0f;
}

// ------------------------------------------------------- histogram of gt>0
// Per sample: 4096 bins over [0,1), each bin keeps (count, sum, sumsq).
__global__ __launch_bounds__(256) void msgil_hist_kernel(const float* __restrict__ gt,
                                                         float* __restrict__ hist) {
    __shared__ float h[3 * NBINS];     // 48 KB
    const int b = blockIdx.y;
    const int t = threadIdx.x;
    for (int i = t; i < 3 * NBINS; i += 256) h[i] = 0.0f;
    __syncthreads();

    const int perChunk = NPIX / CHUNKS;               // 65536
    const size_t base = (size_t)b * NPIX + (size_t)blockIdx.x * perChunk;
    const float4* g4 = (const float4*)(gt + base);
    const int n4 = perChunk / 4;
    for (int i = t; i < n4; i += 256) {
        float4 v = g4[i];
        float vals[4] = {v.x, v.y, v.z, v.w};
        #pragma unroll
        for (int k = 0; k < 4; ++k) {
            float x = vals[k];
            if (x > 0.0f) {
                int bin = (int)(x * (float)NBINS);
                bin = bin < 0 ? 0 : (bin > NBINS - 1 ? NBINS - 1 : bin);
                atomicAdd(&h[bin], 1.0f);               // ds_add_f32
                atomicAdd(&h[NBINS + bin], x);
                atomicAdd(&h[2 * NBINS + bin], x * x);
            }
        }
    }
    __syncthreads();
    float* dst = hist + (size_t)b * 3 * NBINS;
    for (int i = t; i < 3 * NBINS; i += 256)
        if (h[i] != 0.0f) atomicAdd(&dst[i], h[i]);
}

// ------------------------------------------- trimmed mean/std from histogram
// One thread per sample; bins scanned serially (tiny work).
__global__ void msgil_stats_kernel(const float* __restrict__ hist,
                                   float* __restrict__ stats) {
    const int b = threadIdx.x;
    if (b >= BATCH) return;
    const float* hc = hist + (size_t)b * 3 * NBINS;
    const float* hs = hc + NBINS;
    const float* hq = hc + 2 * NBINS;

    float fsize = 0.0f;
    for (int i = 0; i < NBINS; ++i) fsize += hc[i];
    const float trim = floorf(fsize * 0.1f);
    const float lo = trim, hi = fsize - trim;

    float C = 0.0f, s = 0.0f, ss = 0.0f, nk = 0.0f;
    for (int i = 0; i < NBINS; ++i) {
        float cb = hc[i];
        if (cb > 0.0f) {
            float a = C > lo ? C : lo;
            float e = (C + cb) < hi ? (C + cb) : hi;
            float ovl = e - a;
            ovl = ovl < 0.0f ? 0.0f : (ovl > cb ? cb : ovl);
            float f = ovl / cb;
            s  += f * hs[i];
            ss += f * hq[i];
            nk += ovl;
            C  += cb;
        }
    }
    float mean = s / fmaxf(nk, 1.0f);
    float var  = (ss - s * mean) / fmaxf(nk - 1.0f, 1.0f);
    float stdv = sqrtf(fmaxf(var, 0.0f));
    if (fsize < 10.0f) { mean = 0.0f; stdv = 1.0f; }
    stats[2 * b]     = mean;
    stats[2 * b + 1] = stdv;
}

// ------------------------------------------------ gradient loss, one scale
// Tile of the subsampled grid (+2 halo) in LDS; each pixel read once.
__global__ __launch_bounds__(256) void msgil_grad_kernel(const float* __restrict__ pred,
                                                         const float* __restrict__ gt,
                                                         const float* __restrict__ stats,
                                                         float* __restrict__ acc,
                                                         int s, int sIdx) {
    __shared__ float sd[TILE + 2][TILE + 3];
    __shared__ float sm[TILE + 2][TILE + 3];
    __shared__ float red[4][256];

    const int b  = blockIdx.z;
    const int Hs = HDIM / s, Ws = WDIM / s;
    const float mean = stats[2 * b];
    const float inv  = 1.0f / (stats[2 * b + 1] + EPSF);
    const float* P = pred + (size_t)b * NPIX;
    const float* G = gt   + (size_t)b * NPIX;
    const int tx = threadIdx.x, ty = threadIdx.y;
    const int r0 = blockIdx.y * TILE, c0 = blockIdx.x * TILE;

    for (int i = ty; i < TILE + 2; i += 8) {
        const int r = r0 + i;
        for (int j = tx; j < TILE + 2; j += 32) {
            const int c = c0 + j;
            float d = 0.0f, m = 0.0f;
            if (r < Hs && c < Ws) {
                size_t idx = (size_t)(r * s) * WDIM + (size_t)(c * s);
                float p = P[idx];
                float g = G[idx];
                m = (g > VALID_T) ? 1.0f : 0.0f;
                d = p - (g - mean) * inv;
            }
            sd[i][j] = d;
            sm[i][j] = m;
        }
        // prefetch the next row group on the streaming path (global_prefetch_b8)
        int rp = r0 + i + 8; rp = rp < Hs - 1 ? rp : Hs - 1;
        int cp = c0 + tx;    cp = cp < Ws - 1 ? cp : Ws - 1;
        size_t pidx = (size_t)(rp * s) * WDIM + (size_t)(cp * s);
        __builtin_prefetch(&P[pidx], 0, 1);
        __builtin_prefetch(&G[pidx], 0, 1);
    }
    __syncthreads();

    float vs = 0.0f, hs_ = 0.0f, vc = 0.0f, hc = 0.0f;
    for (int i = ty; i < TILE; i += 8) {
        const int r = r0 + i, c = c0 + tx;
        if (r < Hs && c < Ws) {
            const float d0 = sd[i][tx], m0 = sm[i][tx];
            if (r + 2 < Hs) {
                float mm = m0 * sm[i + 2][tx];
                vs += fabsf(d0 - sd[i + 2][tx]) * mm;
                vc += mm;
            }
            if (c + 2 < Ws) {
                float mm = m0 * sm[i][tx + 2];
                hs_ += fabsf(d0 - sd[i][tx + 2]) * mm;
                hc += mm;
            }
        }
    }

    const int t = ty * 32 + tx;
    red[0][t] = vs; red[1][t] = hs_; red[2][t] = vc; red[3][t] = hc;
    __syncthreads();
    for (int st = 128; st > 0; st >>= 1) {
        if (t < st) {
            red[0][t] += red[0][t + st];
            red[1][t] += red[1][t + st];
            red[2][t] += red[2][t + st];
            red[3][t] += red[3][t + st];
        }
        __syncthreads();
    }
    if (t == 0) {
        atomicAdd(&acc[2 * sIdx],     red[0][0] + red[1][0]);  // numerator
        atomicAdd(&acc[2 * sIdx + 1], red[2][0] + red[3][0]);  // masked-pair count
    }
}

// ------------------------------- final combine: K=4 dot product via WMMA f32
// grad_term = sum_s num_s * 1/(den_s + eps). A[0][k]=num_k, B[k][0]=1/(den_k+eps),
// D[0][0] = result (lane 0, first C/D VGPR). 32 threads, EXEC all-ones.
__global__ void msgil_final_kernel(const float* __restrict__ acc,
                                   float* __restrict__ out) {
    const unsigned lane = threadIdx.x;
    float num[4], iv[4];
    #pragma unroll
    for (int i = 0; i < 4; ++i) {
        num[i] = acc[2 * i];
        iv[i]  = 1.0f / (acc[2 * i + 1] + EPSF);
    }
    // A 16x4 f32 layout: VGPR0 lanes0-15 = (M=lane,K=0), lanes16-31 = (M=lane-16,K=2);
    // VGPR1 = K=1 / K=3.  Only row M=0 (lanes 0 and 16) is populated.
    const float l0  = (lane == 0)  ? 1.0f : 0.0f;
    const float l16 = (lane == 16) ? 1.0f : 0.0f;
    v2f a, bb;
    a.x  = l0 * num[0] + l16 * num[2];
    a.y  = l0 * num[1] + l16 * num[3];
    bb.x = l0 * iv[0]  + l16 * iv[2];
    bb.y = l0 * iv[1]  + l16 * iv[3];
    v8f c = {};
    c = __builtin_amdgcn_wmma_f32_16x16x4_f32(false, a, false, bb, (short)0, c,
                                              false, false);
    if (lane == 0) out[0] = c[0];
}

// ----------------------------------------------------------------- launcher
extern "C" void kernel_launch(void* const* d_in, const int* in_sizes, int n_in,
                              void* d_out, int out_size, void* d_ws, size_t ws_size,
                              hipStream_t stream) {
    const float* pred = (const float*)d_in[0];
    const float* gt   = (const float*)d_in[1];
    float* out = (float*)d_out;
    float* ws  = (float*)d_ws;

    float* hist  = ws;                              // 32*3*4096 = 393216 floats
    float* stats = hist + BATCH * 3 * NBINS;        // 64 floats (mean,std)*32
    float* acc   = stats + 2 * BATCH;               // 8 floats (num,den)*4 scales
    const int totalZero = BATCH * 3 * NBINS + 2 * BATCH + 8;

    msgil_zero_kernel<<<(totalZero + 255) / 256, 256, 0, stream>>>(ws, totalZero);

    msgil_hist_kernel<<<dim3(CHUNKS, BATCH), 256, 0, stream>>>(gt, hist);

    msgil_stats_kernel<<<1, 32, 0, stream>>>(hist, stats);

    int s = 1;
    for (int si = 0; si < 4; ++si, s <<= 1) {
        const int Hs = HDIM / s;
        dim3 grid((Hs + TILE - 1) / TILE, (Hs + TILE - 1) / TILE, BATCH);
        msgil_grad_kernel<<<grid, dim3(32, 8), 0, stream>>>(pred, gt, stats, acc, s, si);
    }

    msgil_final_kernel<<<1, 32, 0, stream>>>(acc, out);
}